// BackprojectDepth_18253611008830
// MI455X (gfx1250) — compile-verified
//
#include <hip/hip_runtime.h>

// BackprojectDepth for MI455X (gfx1250, wave32).
// out[b,c,n] = depth[b,n] * (invK[b][c][0]*x + invK[b][c][1]*y + invK[b][c][2]), c=0..2
// out[b,3,n] = 1
// Bandwidth-bound (~335 MB traffic, roofline ~14us @ 23.3 TB/s).
//
// Matrix path: V_WMMA_F32_16X16X4_F32 with K replicated into A rows 0..2 AND
// rows 8..10. D layout (VGPR c: lanes 0-15 = row c, lanes 16-31 = row c+8)
// then gives:
//   WMMA#0 (B = pixels 0..15):  lanes 0-15  of acc0[c] = cam_c(pix 0..15)
//   WMMA#1 (B = pixels 16..31): lanes 16-31 of acc1[c] = cam_c(pix 16..31)
// so a single v_cndmask per channel produces a full-wave coalesced store with
// no cross-lane data movement at all.

typedef __attribute__((ext_vector_type(2))) float v2f;
typedef __attribute__((ext_vector_type(8))) float v8f;

#define B_  32
#define H_  512
#define W_  1024
#define HW_ (H_ * W_)      // 524288
#define ITERS 8            // pixels per block = 256*ITERS = 2048

__global__ __launch_bounds__(256) void backproject_wmma_kernel(
    const float* __restrict__ depth,   // [B,1,H,W]
    const float* __restrict__ invK,    // [B,4,4]
    const int*   __restrict__ dxy,     // [B,2]
    float*       __restrict__ out)     // [B,4,H*W]
{
    const int b    = blockIdx.y;                 // uniform per block -> SGPR
    const int lane = threadIdx.x & 31;
    const int wave = threadIdx.x >> 5;
    const int m    = lane & 15;
    const bool hi  = lane >= 16;

    // Uniform per-batch params (scalarized by compiler).
    const float dxf = (float)dxy[b * 2 + 0];
    const float dyf = (float)dxy[b * 2 + 1];
    const float* K  = invK + b * 16;

    // ---- A matrix (16x4 f32), ISA 32-bit A layout:
    // lane m (0..15): a = (A[m][0], A[m][1]); lane m+16: a = (A[m][2], A[m][3]).
    // Rows 0..2 = invK[:3,:3]; rows 8..10 = invK[:3,:3] (replica); others 0.
    v2f a;
    {
        float e0 = 0.0f, e1 = 0.0f;
        const int r = m & 7;                    // row within half
        if (r < 3) {
            e0 = K[r * 4 + (hi ? 2 : 0)];
            e1 = hi ? 0.0f : K[r * 4 + 1];
        }
        a.x = e0;
        a.y = e1;
    }

    const float* dptr = depth + (size_t)b * HW_;
    float*       ob   = out   + (size_t)b * 4 * HW_;
    const int blockBase = blockIdx.x * (256 * ITERS);

    #pragma unroll
    for (int i = 0; i < ITERS; ++i) {
        const int base32 = blockBase + i * 256 + wave * 32;  // 32-aligned
        const int n      = base32 + lane;                    // output pixel idx

        // Streaming (touch-once) depth read: non-temporal + prefetch ahead.
        const float d = __builtin_nontemporal_load(dptr + n);
        __builtin_prefetch(dptr + n + 256 * ITERS, 0, 1);    // global_prefetch

        // ---- B matrices (4x16 f32) for the two 16-pixel tiles.
        // Lane n (0..15): VGPR0 = B[0][n] = x, VGPR1 = B[1][n] = y.
        // Lanes 16..31:   VGPR0 = B[2][n] = 1, VGPR1 = B[3][n] = 0.
        const int p0 = base32 + m;        // tile 0: pixels base32 + 0..15
        const int p1 = base32 + 16 + m;   // tile 1: pixels base32 + 16..31
        v2f b0, b1;
        b0.x = hi ? 1.0f : (float)(p0 & (W_ - 1)) + dxf;
        b0.y = hi ? 0.0f : (float)(p0 >> 10)      + dyf;
        b1.x = hi ? 1.0f : (float)(p1 & (W_ - 1)) + dxf;
        b1.y = hi ? 0.0f : (float)(p1 >> 10)      + dyf;

        // D[c][n]   = Kc0*x + Kc1*y + Kc2  (rows 0..2)
        // D[c+8][n] = same (rows 8..10, lanes 16..31 of VGPR c)
        v8f acc0 = {};
        v8f acc1 = {};
        acc0 = __builtin_amdgcn_wmma_f32_16x16x4_f32(
                   false, a, false, b0, (short)0, acc0, false, false);
        acc1 = __builtin_amdgcn_wmma_f32_16x16x4_f32(
                   false, a, false, b1, (short)0, acc1, false, false);

        // Lane L<16 -> tile0 cam (acc0[c], row c, N=L);
        // Lane L>=16 -> tile1 cam (acc1[c], row c+8, N=L-16). Pure cndmask.
        #pragma unroll
        for (int c = 0; c < 3; ++c) {
            const float cam = hi ? acc1[c] : acc0[c];
            __builtin_nontemporal_store(cam * d, ob + (size_t)c * HW_ + n);
        }
        __builtin_nontemporal_store(1.0f, ob + (size_t)3 * HW_ + n);
    }
}

extern "C" void kernel_launch(void* const* d_in, const int* in_sizes, int n_in,
                              void* d_out, int out_size, void* d_ws, size_t ws_size,
                              hipStream_t stream) {
    const float* depth = (const float*)d_in[0];   // [32,1,512,1024] f32
    const float* invK  = (const float*)d_in[1];   // [32,4,4] f32
    const int*   dxy   = (const int*)d_in[2];     // [32,2] i32
    float*       out   = (float*)d_out;           // [32,4,524288] f32

    dim3 grid(HW_ / (256 * ITERS), B_);           // (256, 32)
    backproject_wmma_kernel<<<grid, 256, 0, stream>>>(depth, invK, dxy, out);
}